// _ResGCNNet_214748364850
// MI455X (gfx1250) — compile-verified
//
#include <hip/hip_runtime.h>
#include <hip/hip_bf16.h>

typedef __attribute__((ext_vector_type(2))) float v2f;
typedef __attribute__((ext_vector_type(8))) float v8f;

#define NNODES 4096
#define NB     16
#define SCH    128
#define FCH    64
#define KNEI   8
#define LRES   7
#define FUSC   256
#define DCH    1024

#define NTILE 256
#define MTILE 32
#define KTILE 16
#define WSP_LD 96    // per-K-pair stride for W tile: 32*2 + 32 pad (bank shift)
#define XSP_LD 544   // per-K-pair stride for X tile: 256*2 + 32 pad (bank shift)

// ---------------------------------------------------------------------------
// init: zero BN stats slices, zero monotonic-uint max buffer
// ---------------------------------------------------------------------------
__global__ void init_kernel(float* __restrict__ stats, unsigned* __restrict__ gs) {
  int i = blockIdx.x * blockDim.x + threadIdx.x;
  if (i < 8 * 2 * SCH) stats[i] = 0.0f;
  if (i < NB * FUSC)   gs[i] = 0u;        // monotonic map of -inf is > 0, so 0 is identity
}

// ---------------------------------------------------------------------------
// pack [W1 | W2] -> Wcat [O, 2C], bcat = b1 + b2
// ---------------------------------------------------------------------------
__global__ void pack_gcn(const float* __restrict__ w1, const float* __restrict__ w2,
                         const float* __restrict__ b1, const float* __restrict__ b2,
                         float* __restrict__ Wcat, float* __restrict__ bcat,
                         int O, int C) {
  int idx = blockIdx.x * blockDim.x + threadIdx.x;
  int K2 = 2 * C;
  if (idx < O * K2) {
    int o = idx / K2, c = idx % K2;
    Wcat[idx] = (c < C) ? w1[o * C + c] : w2[o * C + (c - C)];
  }
  if (idx < O) bcat[idx] = b1[idx] + b2[idx];
}

// ---------------------------------------------------------------------------
// XN[b, 0:C, n] = src ; XN[b, C:2C, n] = mean_k src[.., adj[n,k]]
// src has batch stride bsSrc, channel stride NNODES
// ---------------------------------------------------------------------------
__global__ void gather_concat(const float* __restrict__ src, long long bsSrc, int C,
                              const int* __restrict__ adj, float* __restrict__ XN) {
  long long idx = (long long)blockIdx.x * blockDim.x + threadIdx.x;
  long long total = (long long)NB * C * NNODES;
  if (idx >= total) return;
  int n = (int)(idx & (NNODES - 1));
  int c = (int)((idx >> 12) % C);
  int b = (int)(idx / ((long long)C * NNODES));
  const float* row = src + (long long)b * bsSrc + (long long)c * NNODES;
  float s = 0.0f;
#pragma unroll
  for (int k = 0; k < KNEI; ++k) s += row[adj[n * KNEI + k]];
  long long xb = (long long)b * (2 * C) * NNODES;
  XN[xb + (long long)c * NNODES + n]       = row[n];
  XN[xb + (long long)(C + c) * NNODES + n] = s * (1.0f / KNEI);
}

// ---------------------------------------------------------------------------
// Generic WMMA fp32 GEMM: Y[b,o,n] = op( sum_k W[o,k]*X[b,k,n] + bias )
//   block = 256 threads = 8 waves, tile 32(M) x 256(N), K staged 16-wide in LDS
//   LDS tiles stored K-pair-interleaved so each WMMA fragment (K=2p, 2p+1)
//   is one aligned ds_load_b64; pads chosen so the two lane-halves of a wave
//   hit disjoint 32-bank groups.
//   modes: relu; stats (per-channel sum/sumsq atomics); maxOut (monotonic uint max)
// ---------------------------------------------------------------------------
__global__ __launch_bounds__(256)
void gemm_wmma(const float* __restrict__ W, int ldW,
               const float* __restrict__ bias,
               const float* __restrict__ bias_bo,   // [B,O] per-batch bias or null
               const float* __restrict__ X, long long bsX,
               float* __restrict__ Y,               // [B,O,N] or null
               float* __restrict__ stats,           // [2*O] or null
               unsigned* __restrict__ maxOut,       // [B,O] or null
               int O, int K, int relu) {
  __shared__ __align__(16) float Wsp[(KTILE / 2) * WSP_LD];
  __shared__ __align__(16) float Xsp[(KTILE / 2) * XSP_LD];

  const int tid   = threadIdx.x;
  const int lane  = tid & 31;
  const int half  = lane >> 4;
  const int l15   = lane & 15;
  const int wave  = tid >> 5;
  const int waveM = wave >> 2;     // 0..1
  const int waveN = wave & 3;      // 0..3
  const int b     = blockIdx.z;
  const int mBase = blockIdx.y * MTILE;
  const int nBase = blockIdx.x * NTILE;

  v8f acc[4] = {};
  const float* Xb = X + (long long)b * bsX;

  for (int k0 = 0; k0 < K; k0 += KTILE) {
    // stage W tile [32 x 16], pair-interleaved (rows past O are zero)
    for (int idx = tid; idx < MTILE * KTILE; idx += 256) {
      int m = idx >> 4, kk = idx & 15;
      int o = mBase + m;
      Wsp[(kk >> 1) * WSP_LD + m * 2 + (kk & 1)] =
          (o < O) ? W[(long long)o * ldW + k0 + kk] : 0.0f;
    }
    // stage X tile [16 x 256], pair-interleaved, global reads coalesced along n
    for (int idx = tid; idx < KTILE * NTILE; idx += 256) {
      int r = idx >> 8, cc = idx & 255;
      Xsp[(r >> 1) * XSP_LD + cc * 2 + (r & 1)] =
          Xb[(long long)(k0 + r) * NNODES + nBase + cc];
    }
    __syncthreads();

#pragma unroll
    for (int kk = 0; kk < KTILE; kk += 4) {
      // pair index for this lane-half: K = kk + 2*half -> p = kk/2 + half
      const int p = (kk >> 1) + half;
      // A fragment: A[M = waveM*16 + l15, K = 2p .. 2p+1]  (ISA 16x4 f32 layout)
      v2f a = *(const v2f*)&Wsp[p * WSP_LD + (waveM * 16 + l15) * 2];
#pragma unroll
      for (int t = 0; t < 4; ++t) {
        int col = waveN * 64 + t * 16 + l15;
        v2f bb = *(const v2f*)&Xsp[p * XSP_LD + col * 2];
        acc[t] = __builtin_amdgcn_wmma_f32_16x16x4_f32(
            false, a, false, bb, (short)0, acc[t], false, false);
      }
    }
    __syncthreads();
  }

  // ---- epilogue ----
  const long long bsY = (long long)O * NNODES;
  float* Yb = Y ? (Y + (long long)b * bsY) : nullptr;

#pragma unroll
  for (int j = 0; j < 8; ++j) {
    int o = mBase + waveM * 16 + j + half * 8;   // C/D layout: M = j + 8*half
    float bv = 0.0f;
    if (o < O) {
      if (bias)    bv  = bias[o];
      if (bias_bo) bv += bias_bo[b * O + o];
    }
    float s = 0.0f, q = 0.0f, mx = -3.402823466e38f;
#pragma unroll
    for (int t = 0; t < 4; ++t) {
      float v = acc[t][j] + bv;
      if (relu) v = fmaxf(v, 0.0f);
      if (Yb && o < O)
        Yb[(long long)o * NNODES + nBase + waveN * 64 + t * 16 + l15] = v;
      s += v; q += v * v; mx = fmaxf(mx, v);
    }
    if (stats || maxOut) {
      // reduce across the 16 lanes of this half (same output row)
      for (int off = 1; off < 16; off <<= 1) {
        s += __shfl_xor(s, off, 32);
        q += __shfl_xor(q, off, 32);
        mx = fmaxf(mx, __shfl_xor(mx, off, 32));
      }
      if (l15 == 0 && o < O) {
        if (stats) {
          atomicAdd(&stats[o], s);
          atomicAdd(&stats[O + o], q);
        }
        if (maxOut) {
          unsigned u = __float_as_uint(mx);
          u = (u & 0x80000000u) ? ~u : (u | 0x80000000u);  // monotonic map
          atomicMax(&maxOut[b * O + o], u);
        }
      }
    }
  }
}

// ---------------------------------------------------------------------------
// BN normalize (+optional skip) from stats; writes into states at dstCh
// ---------------------------------------------------------------------------
__global__ void bn_apply(const float* __restrict__ y, const float* __restrict__ stats,
                         const float* __restrict__ g, const float* __restrict__ bt,
                         float* __restrict__ states, int dstCh, int skipCh) {
  long long idx = (long long)blockIdx.x * blockDim.x + threadIdx.x;
  if (idx >= (long long)NB * SCH * NNODES) return;
  int n = (int)(idx & (NNODES - 1));
  int c = (int)((idx >> 12) & (SCH - 1));
  int b = (int)(idx >> 19);
  const float cnt = (float)NB * (float)NNODES;
  float m   = stats[c] / cnt;
  float var = stats[SCH + c] / cnt - m * m;
  float scale = g[c] / sqrtf(var + 1e-5f);
  float shift = bt[c] - m * scale;
  float v = y[idx] * scale + shift;
  long long sb = (long long)b * DCH * NNODES;
  if (skipCh >= 0) v += states[sb + (long long)(skipCh + c) * NNODES + n];
  states[sb + (long long)(dstCh + c) * NNODES + n] = v;
}

// ---------------------------------------------------------------------------
// per-(b,o) bias for p1 from gs channels: bias_bo = p1_b + p1_w[:,0:256] @ gs[b]
// ---------------------------------------------------------------------------
__global__ void gs_bias(const unsigned* __restrict__ gs_u, const float* __restrict__ p1w,
                        const float* __restrict__ p1b, float* __restrict__ bias_bo) {
  int idx = blockIdx.x * blockDim.x + threadIdx.x;
  if (idx >= NB * FUSC) return;
  int b = idx >> 8, o = idx & 255;
  float s = p1b[o];
  for (int f = 0; f < FUSC; ++f) {
    unsigned u = gs_u[b * FUSC + f];
    u = (u & 0x80000000u) ? (u ^ 0x80000000u) : ~u;   // unmap monotonic
    s += p1w[o * (DCH + FUSC) + f] * __uint_as_float(u);
  }
  bias_bo[idx] = s;
}

// ---------------------------------------------------------------------------
extern "C" void kernel_launch(void* const* d_in, const int* in_sizes, int n_in,
                              void* d_out, int out_size, void* d_ws, size_t ws_size,
                              hipStream_t stream) {
  const float* x       = (const float*)d_in[0];
  const int*   adj     = (const int*)d_in[1];
  const float* head_w1 = (const float*)d_in[2];
  const float* head_b1 = (const float*)d_in[3];
  const float* head_w2 = (const float*)d_in[4];
  const float* head_b2 = (const float*)d_in[5];
  const float* head_g  = (const float*)d_in[6];
  const float* head_bt = (const float*)d_in[7];
  const float* res_w1  = (const float*)d_in[8];
  const float* res_b1  = (const float*)d_in[9];
  const float* res_w2  = (const float*)d_in[10];
  const float* res_b2  = (const float*)d_in[11];
  const float* res_g   = (const float*)d_in[12];
  const float* res_bt  = (const float*)d_in[13];
  const float* fus_w   = (const float*)d_in[14];
  const float* fus_b   = (const float*)d_in[15];
  const float* p1_w    = (const float*)d_in[16];
  const float* p1_b    = (const float*)d_in[17];
  const float* p2_w    = (const float*)d_in[18];
  const float* p2_b    = (const float*)d_in[19];
  const float* p3_w    = (const float*)d_in[20];
  const float* p3_b    = (const float*)d_in[21];

  float* ws = (float*)d_ws;
  long long off = 0;
  float* states = ws + off; off += (long long)NB * DCH * NNODES;        // 256 MB
  float* XN     = ws + off; off += (long long)NB * 2 * SCH * NNODES;    // 64 MB (reused as h1)
  float* tmp    = ws + off; off += (long long)NB * SCH * NNODES;        // 32 MB (reused as h2)
  float* Wcat   = ws + off; off += 8LL * SCH * (2 * SCH);               // 8 slots of 128x256
  float* bcat   = ws + off; off += 8LL * SCH;
  float* stats  = ws + off; off += 8LL * 2 * SCH;
  unsigned* gs  = (unsigned*)(ws + off); off += (long long)NB * FUSC;
  float* biasbo = ws + off; off += (long long)NB * FUSC;
  float* h1 = XN;   // reuse after GCN blocks are done
  float* h2 = tmp;  // reuse after GCN blocks are done

  init_kernel<<<16, 256, 0, stream>>>(stats, gs);

  // pack GCN weights: slot 0 = head (128x128 used), slots 1..7 = res (128x256)
  pack_gcn<<<(SCH * 2 * FCH + 255) / 256, 256, 0, stream>>>(
      head_w1, head_w2, head_b1, head_b2, Wcat, bcat, SCH, FCH);
  for (int i = 0; i < LRES; ++i) {
    pack_gcn<<<(SCH * 2 * SCH + 255) / 256, 256, 0, stream>>>(
        res_w1 + (long long)i * SCH * SCH, res_w2 + (long long)i * SCH * SCH,
        res_b1 + i * SCH, res_b2 + i * SCH,
        Wcat + (long long)(1 + i) * SCH * 2 * SCH, bcat + (1 + i) * SCH, SCH, SCH);
  }

  // ---- head block: F=64 -> S=128, write states[:, 0:128, :] ----
  gather_concat<<<(NB * FCH * NNODES) / 256, 256, 0, stream>>>(
      x, (long long)FCH * NNODES, FCH, adj, XN);
  gemm_wmma<<<dim3(NNODES / NTILE, SCH / MTILE, NB), 256, 0, stream>>>(
      Wcat, 2 * FCH, bcat, nullptr, XN, (long long)2 * FCH * NNODES,
      tmp, stats, nullptr, SCH, 2 * FCH, 1);
  bn_apply<<<(NB * SCH * NNODES) / 256, 256, 0, stream>>>(
      tmp, stats, head_g, head_bt, states, 0, -1);

  // ---- 7 residual blocks ----
  for (int i = 0; i < LRES; ++i) {
    gather_concat<<<(NB * SCH * NNODES) / 256, 256, 0, stream>>>(
        states + (long long)i * SCH * NNODES, (long long)DCH * NNODES, SCH, adj, XN);
    gemm_wmma<<<dim3(NNODES / NTILE, SCH / MTILE, NB), 256, 0, stream>>>(
        Wcat + (long long)(1 + i) * SCH * 2 * SCH, 2 * SCH,
        bcat + (1 + i) * SCH, nullptr, XN, (long long)2 * SCH * NNODES,
        tmp, stats + (1 + i) * 2 * SCH, nullptr, SCH, 2 * SCH, 1);
    bn_apply<<<(NB * SCH * NNODES) / 256, 256, 0, stream>>>(
        tmp, stats + (1 + i) * 2 * SCH, res_g + i * SCH, res_bt + i * SCH,
        states, (i + 1) * SCH, i * SCH);
  }

  // ---- fusion: gs[b,o] = max_n (fus_w @ state + fus_b) ----
  gemm_wmma<<<dim3(NNODES / NTILE, FUSC / MTILE, NB), 256, 0, stream>>>(
      fus_w, DCH, fus_b, nullptr, states, (long long)DCH * NNODES,
      nullptr, nullptr, gs, FUSC, DCH, 0);

  // ---- fold gs channels of p1 into a per-(b,o) bias ----
  gs_bias<<<NB, 256, 0, stream>>>(gs, p1_w, p1_b, biasbo);

  // ---- p1: h1 = relu(p1_w[:,256:1280] @ state + biasbo) ----
  gemm_wmma<<<dim3(NNODES / NTILE, FUSC / MTILE, NB), 256, 0, stream>>>(
      p1_w + FUSC, DCH + FUSC, nullptr, biasbo, states, (long long)DCH * NNODES,
      h1, nullptr, nullptr, FUSC, DCH, 1);

  // ---- p2: h2 = relu(p2_w @ h1 + p2_b) ----
  gemm_wmma<<<dim3(NNODES / NTILE, 64 / MTILE, NB), 256, 0, stream>>>(
      p2_w, 256, p2_b, nullptr, h1, (long long)256 * NNODES,
      h2, nullptr, nullptr, 64, 256, 1);

  // ---- p3: out = p3_w @ h2 + p3_b  (O=2, masked M tile) ----
  gemm_wmma<<<dim3(NNODES / NTILE, 1, NB), 256, 0, stream>>>(
      p3_w, 64, p3_b, nullptr, h2, (long long)64 * NNODES,
      (float*)d_out, nullptr, nullptr, 2, 64, 0);
}